// InterModalRankingLoss_14998025798301
// MI455X (gfx1250) — compile-verified
//
#include <hip/hip_runtime.h>
#include <math.h>

#define BDIM 4096
#define DDIM 1024
#define TOPK 1024u
#define LSTR 20          // LDS row stride (floats): bank-conflict padding, 8B aligned

typedef float v2f __attribute__((ext_vector_type(2)));
typedef float v8f __attribute__((ext_vector_type(8)));

#if defined(__AMDGCN__) && __has_builtin(__builtin_amdgcn_global_load_async_to_lds_b128) && __has_builtin(__builtin_amdgcn_s_wait_asynccnt)
#define USE_ASYNC_LDS 1
typedef int v4i __attribute__((vector_size(16)));            // matches builtin param type
typedef __attribute__((address_space(1))) v4i gv4i;          // global (device) src
typedef __attribute__((address_space(3))) v4i sv4i;          // LDS (shared) dst
#else
#define USE_ASYNC_LDS 0
#endif

// ---------------- row norms / sums ----------------
__global__ __launch_bounds__(256)
void norms_kernel(const float* __restrict__ x, const float* __restrict__ y,
                  float* __restrict__ nx, float* __restrict__ sx,
                  float* __restrict__ ny, float* __restrict__ sy)
{
    const int row  = blockIdx.x * 8 + (threadIdx.x >> 5);
    const int lane = threadIdx.x & 31;
    float sxx = 0.f, sxs = 0.f, syy = 0.f, sys = 0.f;
    for (int j = lane; j < DDIM; j += 32) {
        float xv = x[(size_t)row * DDIM + j];
        float yv = y[(size_t)row * DDIM + j];
        sxx += xv * xv; sxs += xv;
        syy += yv * yv; sys += yv;
    }
    #pragma unroll
    for (int m = 16; m > 0; m >>= 1) {
        sxx += __shfl_xor(sxx, m, 32);
        sxs += __shfl_xor(sxs, m, 32);
        syy += __shfl_xor(syy, m, 32);
        sys += __shfl_xor(sys, m, 32);
    }
    if (lane == 0) { nx[row] = sxx; sx[row] = sxs; ny[row] = syy; sy[row] = sys; }
}

// ---------------- fp32 WMMA distance GEMM (async double-buffered LDS) ----------------
__global__ __launch_bounds__(256)
void dist_gemm_kernel(const float* __restrict__ x, const float* __restrict__ y,
                      const float* __restrict__ nx, const float* __restrict__ sx,
                      const float* __restrict__ ny, const float* __restrict__ sy,
                      float* __restrict__ dist, float* __restrict__ distT)
{
    __shared__ float ldsX[2][128 * LSTR];
    __shared__ float ldsY[2][128 * LSTR];
    __shared__ float ldsT[8 * 16 * 17];     // per-wave transpose staging

    const int tid    = threadIdx.x;
    const int lane   = tid & 31;
    const int waveId = tid >> 5;
    const int rowBase = blockIdx.y * 128;
    const int colBase = blockIdx.x * 128;
    const int wRow = (waveId >> 2) * 64;    // 2 wave-rows x 4 wave-cols
    const int wCol = (waveId & 3) * 32;
    const int mlan  = lane & 15;
    const int half  = lane >> 4;
    const int khalf = half * 2;

    v8f acc[4][2];
    #pragma unroll
    for (int r = 0; r < 4; ++r)
        #pragma unroll
        for (int c = 0; c < 2; ++c)
            #pragma unroll
            for (int e = 0; e < 8; ++e) acc[r][c][e] = 0.0f;

    auto compute_stage = [&](int buf) {
        #pragma unroll
        for (int kk = 0; kk < 16; kk += 4) {
            v2f a[4], b[2];
            #pragma unroll
            for (int r = 0; r < 4; ++r)
                a[r] = *(const v2f*)(&ldsX[buf][(wRow + r * 16 + mlan) * LSTR + kk + khalf]);
            #pragma unroll
            for (int c = 0; c < 2; ++c)
                b[c] = *(const v2f*)(&ldsY[buf][(wCol + c * 16 + mlan) * LSTR + kk + khalf]);
            #pragma unroll
            for (int r = 0; r < 4; ++r)
                #pragma unroll
                for (int c = 0; c < 2; ++c)
                    acc[r][c] = __builtin_amdgcn_wmma_f32_16x16x4_f32(
                        false, a[r], false, b[c], (short)0, acc[r][c], false, false);
        }
    };

#if USE_ASYNC_LDS
    // Async global->LDS copy engine path (ASYNCcnt), 4 b128 issues per thread per stage.
    auto issue_stage = [&](int k0, int buf) {
        #pragma unroll
        for (int q = 0; q < 2; ++q) {
            int idx = q * 256 + tid;
            int r   = idx >> 2;
            int f   = (idx & 3) * 4;
            __builtin_amdgcn_global_load_async_to_lds_b128(
                (gv4i*)(x + (size_t)(rowBase + r) * DDIM + k0 + f),
                (sv4i*)&ldsX[buf][r * LSTR + f], 0, 0);
            __builtin_amdgcn_global_load_async_to_lds_b128(
                (gv4i*)(y + (size_t)(colBase + r) * DDIM + k0 + f),
                (sv4i*)&ldsY[buf][r * LSTR + f], 0, 0);
        }
    };
    issue_stage(0, 0);
    for (int k0 = 0; k0 < DDIM; k0 += 16) {
        const int cur = (k0 >> 4) & 1;
        if (k0 + 16 < DDIM) {
            issue_stage(k0 + 16, cur ^ 1);          // prefetch next stage
            __builtin_amdgcn_s_wait_asynccnt(4);    // wait only for current stage
        } else {
            __builtin_amdgcn_s_wait_asynccnt(0);
        }
        __syncthreads();                            // current stage visible to all waves
        compute_stage(cur);
        __syncthreads();                            // all reads done before buffer reuse
    }
#else
    for (int k0 = 0; k0 < DDIM; k0 += 16) {
        __syncthreads();
        #pragma unroll
        for (int q = 0; q < 2; ++q) {               // 512 float4 loads by 256 threads
            int idx = q * 256 + tid;
            int r   = idx >> 2;
            int f   = (idx & 3) * 4;
            float4 vx = *(const float4*)(x + (size_t)(rowBase + r) * DDIM + k0 + f);
            float4 vy = *(const float4*)(y + (size_t)(colBase + r) * DDIM + k0 + f);
            *(float4*)(&ldsX[0][r * LSTR + f]) = vx;
            *(float4*)(&ldsY[0][r * LSTR + f]) = vy;
            if (k0 + 16 < DDIM) {
                __builtin_prefetch(x + (size_t)(rowBase + r) * DDIM + k0 + 16 + f, 0, 3);
                __builtin_prefetch(y + (size_t)(colBase + r) * DDIM + k0 + 16 + f, 0, 3);
            }
        }
        __syncthreads();
        compute_stage(0);
    }
#endif

    // Epilogue: sq = ||x||^2 + ||y||^2 - 2 x.y + 2*eps*(sx - sy) + D*eps^2
    const int wb = waveId * 16 * 17;
    #pragma unroll
    for (int r = 0; r < 4; ++r) {
        #pragma unroll
        for (int c = 0; c < 2; ++c) {
            const int gr0 = rowBase + wRow + r * 16;
            const int gc0 = colBase + wCol + c * 16;
            const int gj  = gc0 + mlan;                 // N = lane&15
            float dv[8];
            #pragma unroll
            for (int v = 0; v < 8; ++v) {
                const int gi = gr0 + v + 8 * half;      // M = v + 8*(lane>>4)
                float sq = nx[gi] + ny[gj] - 2.0f * acc[r][c][v]
                         + 2.0e-6f * (sx[gi] - sy[gj]) + 1.024e-9f;
                float d = sqrtf(fmaxf(sq, 0.0f));
                dist[(size_t)gi * BDIM + gj] = d;       // coalesced row store
                dv[v] = d;
            }
            // wave-private LDS transpose -> coalesced distT store (same-wave DS is in-order)
            #pragma unroll
            for (int v = 0; v < 8; ++v)
                ldsT[wb + mlan * 17 + (v + 8 * half)] = dv[v];
            #pragma unroll
            for (int v = 0; v < 8; ++v) {
                const int i0 = v + 8 * half;
                float tval = ldsT[wb + i0 * 17 + mlan];
                distT[(size_t)(gc0 + i0) * BDIM + (gr0 + mlan)] = tval;
            }
        }
    }
}

// ---------------- per-row radix-select + hinge sum ----------------
__global__ __launch_bounds__(256)
void topk_loss_kernel(const float* __restrict__ dist, const float* __restrict__ distT,
                      float* __restrict__ rowloss)
{
    __shared__ unsigned keys[BDIM];          // 16 KB
    __shared__ unsigned whist[8 * 256];      // per-wave histograms, 8 KB
    __shared__ unsigned scan[256];
    __shared__ float    fsum[256];
    __shared__ unsigned s_prefix, s_t;

    const int tid  = threadIdx.x;
    const int wave = tid >> 5;
    const int bid  = blockIdx.x;
    const int row  = bid & (BDIM - 1);
    const float* mat = (bid < BDIM) ? dist : distT;

    const float pos = mat[(size_t)row * BDIM + row];

    for (int i = tid; i < BDIM; i += 256) {
        float d = __builtin_nontemporal_load(mat + (size_t)row * BDIM + i);
        if (i == row) d = 1.0e6f;                    // mask diagonal with BIG
        keys[i] = __float_as_uint(d);                // d >= 0 -> order-preserving
    }
    if (tid == 0) { s_prefix = 0u; s_t = TOPK; }
    __syncthreads();

    // k-th smallest key: 4 x 8-bit MSB-first radix select, per-wave histograms
    for (int pass = 0; pass < 4; ++pass) {
        const int shift = 24 - 8 * pass;
        const unsigned prefix = s_prefix;
        const unsigned tneed  = s_t;
        #pragma unroll
        for (int w = 0; w < 8; ++w) whist[w * 256 + tid] = 0u;
        __syncthreads();
        for (int i = tid; i < BDIM; i += 256) {
            unsigned kkey = keys[i];
            bool match = (pass == 0) || (((kkey ^ prefix) >> (shift + 8)) == 0u);
            if (match) atomicAdd(&whist[wave * 256 + ((kkey >> shift) & 255u)], 1u);
        }
        __syncthreads();
        unsigned cnt = 0u;                           // this thread owns bin `tid`
        #pragma unroll
        for (int w = 0; w < 8; ++w) cnt += whist[w * 256 + tid];
        scan[tid] = cnt;
        __syncthreads();
        #pragma unroll
        for (int off = 1; off < 256; off <<= 1) {    // inclusive Hillis-Steele scan
            unsigned v = (tid >= off) ? scan[tid - off] : 0u;
            __syncthreads();
            scan[tid] += v;
            __syncthreads();
        }
        const unsigned incl = scan[tid];
        const unsigned excl = incl - cnt;
        if (excl < tneed && incl >= tneed) {         // exactly one bin owner fires
            s_prefix = prefix | ((unsigned)tid << shift);
            s_t = tneed - excl;
        }
        __syncthreads();
    }

    const unsigned T = s_prefix;                     // exact k-th smallest key
    float lsum = 0.0f; unsigned lcnt = 0u;
    for (int i = tid; i < BDIM; i += 256) {
        unsigned kkey = keys[i];
        if (kkey < T) {
            lsum += fmaxf(0.0f, 0.5f + pos - __uint_as_float(kkey));
            lcnt++;
        }
    }
    fsum[tid] = lsum; scan[tid] = lcnt;
    __syncthreads();
    #pragma unroll
    for (int st = 128; st > 0; st >>= 1) {
        if (tid < st) { fsum[tid] += fsum[tid + st]; scan[tid] += scan[tid + st]; }
        __syncthreads();
    }
    if (tid == 0) {
        float hT = fmaxf(0.0f, 0.5f + pos - __uint_as_float(T));
        rowloss[bid] = fsum[0] + (float)(TOPK - scan[0]) * hT;   // ties at threshold
    }
}

// ---------------- deterministic final reduce ----------------
__global__ __launch_bounds__(256)
void reduce_kernel(const float* __restrict__ rowloss, float* __restrict__ out)
{
    __shared__ float sh[256];
    const int b = blockIdx.x;                         // 0 = loss_xy, 1 = loss_yx
    float s = 0.0f;
    for (int i = threadIdx.x; i < BDIM; i += 256) s += rowloss[b * BDIM + i];
    sh[threadIdx.x] = s;
    __syncthreads();
    #pragma unroll
    for (int st = 128; st > 0; st >>= 1) {
        if (threadIdx.x < st) sh[threadIdx.x] += sh[threadIdx.x + st];
        __syncthreads();
    }
    if (threadIdx.x == 0)
        out[b] = sh[0] * (1.0f / ((float)BDIM * (float)TOPK));
}

extern "C" void kernel_launch(void* const* d_in, const int* in_sizes, int n_in,
                              void* d_out, int out_size, void* d_ws, size_t ws_size,
                              hipStream_t stream)
{
    (void)in_sizes; (void)n_in; (void)out_size; (void)ws_size;
    const float* x = (const float*)d_in[0];
    const float* y = (const float*)d_in[1];
    float* out = (float*)d_out;
    float* ws  = (float*)d_ws;

    float* dist  = ws;
    float* distT = dist  + (size_t)BDIM * BDIM;
    float* nx    = distT + (size_t)BDIM * BDIM;
    float* sx    = nx + BDIM;
    float* ny    = sx + BDIM;
    float* sy    = ny + BDIM;
    float* rowloss = sy + BDIM;

    norms_kernel<<<BDIM / 8, 256, 0, stream>>>(x, y, nx, sx, ny, sy);
    dist_gemm_kernel<<<dim3(BDIM / 128, BDIM / 128), 256, 0, stream>>>(
        x, y, nx, sx, ny, sy, dist, distT);
    topk_loss_kernel<<<2 * BDIM, 256, 0, stream>>>(dist, distT, rowloss);
    reduce_kernel<<<2, 256, 0, stream>>>(rowloss, out);
}